// TransformerGrammar_40364102648071
// MI455X (gfx1250) — compile-verified
//
#include <hip/hip_runtime.h>

// ---------------------------------------------------------------------------
// Transformer-XL layer for MI455X (gfx1250), fp32 end-to-end using
// V_WMMA_F32_16X16X4_F32 for every GEMM-shaped computation.
// ---------------------------------------------------------------------------

typedef float f32x2 __attribute__((ext_vector_type(2)));
typedef float f32x8 __attribute__((ext_vector_type(8)));

#define TG_QLEN 1024
#define TG_BSZ  4
#define TG_NH   10
#define TG_DH   38
#define TG_DM   380
#define TG_RLEN 2049
#define TG_HROW 1140           // 3*NH*DH, row stride of fused qkv "heads"
#define TG_SCALE 0.16222142113076254f  // 1/sqrt(38)

static __device__ __forceinline__ f32x8 wmma4(f32x2 a, f32x2 b, f32x8 c) {
  // D(16x16,f32) = A(16x4,f32) x B(4x16,f32) + C  -- full fp32 matrix op
  return __builtin_amdgcn_wmma_f32_16x16x4_f32(
      /*neg_a=*/false, a, /*neg_b=*/false, b,
      /*c_mod=*/(short)0, c, /*reuse_a=*/false, /*reuse_b=*/false);
}

// ---------------------------------------------------------------------------
// Generic C[M,N] = A[M,K] * B[N,K]^T (+bias, +relu).
// One wave computes a 16x64 tile.  The inner loop processes 4 K-steps (64 B)
// per body: one address per pointer serves 4 immediate-offset loads, so the
// 20 loads batch into clauses and the 16 WMMAs consume them with a
// descending loadcnt wait chain (not loadcnt==0 per WMMA).
// Edge tiles: loads are CLAMPED (branchless inner loop); stores are guarded.
// K must be a multiple of 4 (true for 380 and 900).
// ---------------------------------------------------------------------------
__global__ __launch_bounds__(128) void gemm_nt_wmma(
    const float* __restrict__ A, int lda,
    const float* __restrict__ B, int ldb,
    float* __restrict__ C, int ldc,
    int M, int N, int K,
    const float* __restrict__ bias, int relu)
{
  const int lane = threadIdx.x & 31;
  const int wid  = threadIdx.x >> 5;
  const int tiles_n = (N + 63) >> 6;              // 64-wide N tiles
  const int tiles_m = (M + 15) >> 4;
  const int tile = blockIdx.x * 4 + wid;
  if (tile >= tiles_m * tiles_n) return;          // uniform per wave
  const int tm = tile / tiles_n;
  const int tn = tile % tiles_n;

  const int mr   = lane & 15;       // fragment row/col within tile
  const int half = lane >> 4;
  const int koff = half * 2;

  const int arow = min(tm * 16 + mr, M - 1);      // clamped (store-guarded)
  const float* Ap = A + (size_t)arow * lda + koff;
  int ncol[4];
#pragma unroll
  for (int c = 0; c < 4; ++c) ncol[c] = tn * 64 + c * 16 + mr;
  const float* Bp0 = B + (size_t)min(ncol[0], N - 1) * ldb + koff;
  const float* Bp1 = B + (size_t)min(ncol[1], N - 1) * ldb + koff;
  const float* Bp2 = B + (size_t)min(ncol[2], N - 1) * ldb + koff;
  const float* Bp3 = B + (size_t)min(ncol[3], N - 1) * ldb + koff;

  f32x8 acc0 = {0.f,0.f,0.f,0.f,0.f,0.f,0.f,0.f};
  f32x8 acc1 = acc0, acc2 = acc0, acc3 = acc0;

  int k0 = 0;
#pragma unroll 1
  for (; k0 + 16 <= K; k0 += 16) {
    // 20 independent loads, 4 immediate offsets per address -> load clauses
    f32x2 a[4], b0[4], b1[4], b2[4], b3[4];
#pragma unroll
    for (int u = 0; u < 4; ++u) {
      a[u]  = *(const f32x2*)(Ap  + k0 + 4 * u);
      b0[u] = *(const f32x2*)(Bp0 + k0 + 4 * u);
      b1[u] = *(const f32x2*)(Bp1 + k0 + 4 * u);
      b2[u] = *(const f32x2*)(Bp2 + k0 + 4 * u);
      b3[u] = *(const f32x2*)(Bp3 + k0 + 4 * u);
    }
#pragma unroll
    for (int u = 0; u < 4; ++u) {
      acc0 = wmma4(a[u], b0[u], acc0);
      acc1 = wmma4(a[u], b1[u], acc1);
      acc2 = wmma4(a[u], b2[u], acc2);
      acc3 = wmma4(a[u], b3[u], acc3);
    }
  }
#pragma unroll 1
  for (; k0 < K; k0 += 4) {                       // K%16 tail (1-3 steps)
    const f32x2 a0  = *(const f32x2*)(Ap  + k0);
    const f32x2 b00 = *(const f32x2*)(Bp0 + k0);
    const f32x2 b01 = *(const f32x2*)(Bp1 + k0);
    const f32x2 b02 = *(const f32x2*)(Bp2 + k0);
    const f32x2 b03 = *(const f32x2*)(Bp3 + k0);
    acc0 = wmma4(a0, b00, acc0);
    acc1 = wmma4(a0, b01, acc1);
    acc2 = wmma4(a0, b02, acc2);
    acc3 = wmma4(a0, b03, acc3);
  }

  f32x8 accs[4] = {acc0, acc1, acc2, acc3};
#pragma unroll
  for (int c = 0; c < 4; ++c) {
    const int n = ncol[c];
    if (n >= N) continue;
    const float bv = (bias != nullptr) ? bias[n] : 0.f;
#pragma unroll
    for (int r = 0; r < 8; ++r) {
      const int m = tm * 16 + r + half * 8;       // D layout: M = r + 8*(lane/16)
      if (m < M) {
        float v = accs[c][r] + bv;
        if (relu) v = fmaxf(v, 0.f);
        C[(size_t)m * ldc + n] = v;
      }
    }
  }
}

// ---------------------------------------------------------------------------
// Fused rel-attention per (16-query block, head, batch):
//   1) BDrow[16][2049] = (q+r_r_bias) @ rk_head^T   (WMMA, into LDS)
//   2) S[16][1024]     = (q+r_w_bias) @ k^T  + gather(BDrow, idx), mask, scale
//   3) row softmax in LDS
//   4) vec[16][38]     = P @ V                      (WMMA)
// A fragments (q+bias, zero-padded K 38->40) are hoisted into registers; the
// 10 B fragments of each column tile are batch-loaded before the WMMA chain.
// LDS: 2*16*40 + 16*2052 + 16*1026 floats = ~202 KB (< 320 KB/WGP on CDNA5)
// ---------------------------------------------------------------------------
__global__ __launch_bounds__(256) void attn_kernel(
    const float* __restrict__ heads,   // [4096][1140]  (q | k | v)
    const float* __restrict__ rk,      // [2049][380] (+16 floats slack after)
    const float* __restrict__ rwb,     // [10][38]
    const float* __restrict__ rrb,     // [10][38]
    const int*  __restrict__ relpos,   // [4][1024][1024]
    const unsigned char* __restrict__ mask, // [1024][1024][4] (bool)
    float* __restrict__ vec)           // [4096][380]
{
  __shared__ float qw[16][40];         // q + r_w_bias, K padded 38->40 (zeros)
  __shared__ float qr[16][40];         // q + r_r_bias
  __shared__ float bd[16][2052];       // full relative-position row per query
  __shared__ float sbuf[16][1026];     // scores -> probabilities

  const int i0 = blockIdx.x * 16;
  const int n  = blockIdx.y;
  const int b  = blockIdx.z;
  const int t  = threadIdx.x;
  const int lane = t & 31;
  const int wid  = t >> 5;             // 8 waves
  const int mr   = lane & 15;
  const int half = lane >> 4;
  const int koff = half * 2;

  // ---- stage q with biases (zero-padded K) --------------------------------
  for (int e = t; e < 16 * 40; e += 256) {
    const int ii = e / 40, d = e % 40;
    float qv = 0.f, w1 = 0.f, w2 = 0.f;
    if (d < TG_DH) {
      const int row = (i0 + ii) * TG_BSZ + b;
      qv = heads[(size_t)row * TG_HROW + n * TG_DH + d];
      w1 = rwb[n * TG_DH + d];
      w2 = rrb[n * TG_DH + d];
    }
    qw[ii][d] = qv + w1;
    qr[ii][d] = qv + w2;
  }
  __syncthreads();

  // ---- hoist A fragments: invariant across all column tiles ---------------
  f32x2 aR[10], aW[10];
#pragma unroll
  for (int ks = 0; ks < 10; ++ks) {
    aR[ks] = *(const f32x2*)&qr[mr][ks * 4 + koff];
    aW[ks] = *(const f32x2*)&qw[mr][ks * 4 + koff];
  }

  // ---- phase 1: BDrow = qr @ rk_head^T  (129 column tiles over 8 waves) ---
  // A cols 38/39 are zero, so unguarded B loads at d=38/39 contribute exactly
  // 0 (rk region has tail slack, values finite).
#pragma unroll 1
  for (int jt = wid; jt < (TG_RLEN + 15) / 16; jt += 8) {
    const int jr  = jt * 16 + mr;               // B-fragment: N = lane%16
    const int jrl = min(jr, TG_RLEN - 1);       // clamped load row
    const float* Bp = rk + (size_t)jrl * TG_DM + n * TG_DH + koff;
    f32x2 bb[10];
#pragma unroll
    for (int ks = 0; ks < 10; ++ks) bb[ks] = *(const f32x2*)(Bp + ks * 4);
    f32x8 acc = {0.f,0.f,0.f,0.f,0.f,0.f,0.f,0.f};
#pragma unroll
    for (int ks = 0; ks < 10; ++ks) acc = wmma4(aR[ks], bb[ks], acc);
    if (jr < TG_RLEN) {
#pragma unroll
      for (int r = 0; r < 8; ++r) bd[r + half * 8][jr] = acc[r];
    }
  }
  __syncthreads();

  // ---- phase 2: scores = AC + gathered BD, mask, scale --------------------
#pragma unroll 1
  for (int jt = wid; jt < TG_QLEN / 16; jt += 8) {
    const int j = jt * 16 + mr;                 // key index (B-fragment N)
    const float* Bp = heads + (size_t)(j * TG_BSZ + b) * TG_HROW + TG_DM + n * TG_DH + koff;
    f32x2 bb[10];
#pragma unroll
    for (int ks = 0; ks < 10; ++ks) bb[ks] = *(const f32x2*)(Bp + ks * 4);
    f32x8 acc = {0.f,0.f,0.f,0.f,0.f,0.f,0.f,0.f};
#pragma unroll
    for (int ks = 0; ks < 10; ++ks) acc = wmma4(aW[ks], bb[ks], acc);
#pragma unroll
    for (int r = 0; r < 8; ++r) {
      const int ii = r + half * 8;
      const int i = i0 + ii;
      int rp = relpos[((size_t)b * TG_QLEN + i) * TG_QLEN + j];
      rp = min(1024, max(-1024, rp));
      const int jdx = 1024 - rp;                // gather index into BDrow
      float s;
      if (mask[((size_t)i * TG_QLEN + j) * TG_BSZ + b])
        s = (acc[r] + bd[ii][jdx]) * TG_SCALE;
      else
        s = -1e30f;
      sbuf[ii][j] = s;
    }
  }
  __syncthreads();

  // ---- phase 3: row softmax (16 lanes per row, shfl_xor reductions) -------
  {
    const int row = t >> 4;            // 0..15
    const int c   = t & 15;
    float m = -1e30f;
    for (int j = c; j < TG_QLEN; j += 16) m = fmaxf(m, sbuf[row][j]);
#pragma unroll
    for (int off = 1; off < 16; off <<= 1) m = fmaxf(m, __shfl_xor(m, off, 32));
    float s = 0.f;
    for (int j = c; j < TG_QLEN; j += 16) {
      const float e = __expf(sbuf[row][j] - m);
      sbuf[row][j] = e;
      s += e;
    }
#pragma unroll
    for (int off = 1; off < 16; off <<= 1) s += __shfl_xor(s, off, 32);
    const float inv = 1.f / s;
    for (int j = c; j < TG_QLEN; j += 16) sbuf[row][j] *= inv;
  }
  __syncthreads();

  // ---- phase 4: vec = P @ V  (3 column tiles of d, one wave each) ---------
  if (wid < 3) {
    const int d  = wid * 16 + mr;      // B-fragment N = value dim
    const int dl = min(d, TG_DH - 1);  // clamped load col (store-guarded)
    const float* Vp = heads + 2 * TG_DM + n * TG_DH + dl + (size_t)b * TG_HROW;
    f32x8 acc = {0.f,0.f,0.f,0.f,0.f,0.f,0.f,0.f};
    // batch 4 K-steps: 8 strided V loads + 4 LDS reads, then 4 WMMAs
#pragma unroll 1
    for (int k0 = 0; k0 < TG_QLEN; k0 += 16) {
      f32x2 a[4], bb[4];
#pragma unroll
      for (int u = 0; u < 4; ++u) {
        const int j = k0 + u * 4 + koff;
        bb[u].x = Vp[(size_t) j      * (TG_BSZ * TG_HROW)];
        bb[u].y = Vp[(size_t)(j + 1) * (TG_BSZ * TG_HROW)];
        a[u] = *(const f32x2*)&sbuf[mr][j];
      }
#pragma unroll
      for (int u = 0; u < 4; ++u) acc = wmma4(a[u], bb[u], acc);
    }
    if (d < TG_DH) {
#pragma unroll
      for (int r = 0; r < 8; ++r)
        vec[(size_t)((i0 + r + half * 8) * TG_BSZ + b) * TG_DM + n * TG_DH + d] = acc[r];
    }
  }
}

// ---------------------------------------------------------------------------
// out[row] = LayerNorm(x[row] + y[row]) * g + b    (D = 380, 128 threads/row)
// ---------------------------------------------------------------------------
__global__ __launch_bounds__(128) void add_ln_kernel(
    const float* __restrict__ x, const float* __restrict__ y,
    const float* __restrict__ g, const float* __restrict__ bta,
    float* __restrict__ out, int D)
{
  const int row = blockIdx.x;
  const int t = threadIdx.x;
  __shared__ float part[4];

  float v[3]; int cnt = 0;
  float s = 0.f;
  for (int d = t; d < D; d += 128) {
    const float vv = x[(size_t)row * D + d] + y[(size_t)row * D + d];
    v[cnt++] = vv; s += vv;
  }
#pragma unroll
  for (int off = 16; off > 0; off >>= 1) s += __shfl_xor(s, off, 32);
  if ((t & 31) == 0) part[t >> 5] = s;
  __syncthreads();
  const float mean = (part[0] + part[1] + part[2] + part[3]) / (float)D;
  __syncthreads();

  float vs = 0.f;
  for (int k = 0; k < cnt; ++k) { const float d0 = v[k] - mean; vs += d0 * d0; }
#pragma unroll
  for (int off = 16; off > 0; off >>= 1) vs += __shfl_xor(vs, off, 32);
  if ((t & 31) == 0) part[t >> 5] = vs;
  __syncthreads();
  const float var = (part[0] + part[1] + part[2] + part[3]) / (float)D;
  const float inv = rsqrtf(var + 1e-5f);

  cnt = 0;
  for (int d = t; d < D; d += 128)
    out[(size_t)row * D + d] = (v[cnt++] - mean) * inv * g[d] + bta[d];
}

// ---------------------------------------------------------------------------
extern "C" void kernel_launch(void* const* d_in, const int* in_sizes, int n_in,
                              void* d_out, int out_size, void* d_ws, size_t ws_size,
                              hipStream_t stream) {
  (void)in_sizes; (void)n_in; (void)out_size; (void)ws_size;
  const float* w    = (const float*)d_in[0];
  const float* r    = (const float*)d_in[1];
  const float* rwb  = (const float*)d_in[2];
  const float* rrb  = (const float*)d_in[3];
  const float* qkvw = (const float*)d_in[4];
  const float* r_w  = (const float*)d_in[5];
  const float* o_w  = (const float*)d_in[6];
  const float* ln1g = (const float*)d_in[7];
  const float* ln1b = (const float*)d_in[8];
  const float* ffw1 = (const float*)d_in[9];
  const float* ffb1 = (const float*)d_in[10];
  const float* ffw2 = (const float*)d_in[11];
  const float* ffb2 = (const float*)d_in[12];
  const float* ln2g = (const float*)d_in[13];
  const float* ln2b = (const float*)d_in[14];
  const unsigned char* mask = (const unsigned char*)d_in[15];
  const int* relpos = (const int*)d_in[16];
  float* out = (float*)d_out;

  // workspace layout (fp32), total ~61.5 MB
  float* ws       = (float*)d_ws;
  float* heads    = ws;                               // 4096 x 1140
  float* rk       = heads    + 4096 * 1140;           // 2049 x 380 (+16 slack)
  float* vec      = rk       + 2049 * 380 + 16;       // 4096 x 380
  float* attn_out = vec      + 4096 * 380;            // 4096 x 380
  float* out1     = attn_out + 4096 * 380;            // 4096 x 380
  float* h1       = out1     + 4096 * 380;            // 4096 x 900
  float* core     = h1       + 4096 * 900;            // 4096 x 380

  auto gemm = [&](const float* A, int lda, const float* B, int ldb,
                  float* C, int ldc, int M, int N, int K,
                  const float* bias, int relu) {
    const int tiles = ((M + 15) / 16) * ((N + 63) / 64);
    const int blocks = (tiles + 3) / 4;
    gemm_nt_wmma<<<blocks, 128, 0, stream>>>(A, lda, B, ldb, C, ldc, M, N, K, bias, relu);
  };

  // 1) fused qkv projection: heads = w @ qkv_w^T
  gemm(w, 380, qkvw, 380, heads, 1140, 4096, 1140, 380, nullptr, 0);
  // 2) rk = r @ r_w^T
  gemm(r, 380, r_w, 380, rk, 380, 2049, 380, 380, nullptr, 0);
  // 3) fused relative attention -> vec
  attn_kernel<<<dim3(TG_QLEN / 16, TG_NH, TG_BSZ), 256, 0, stream>>>(
      heads, rk, rwb, rrb, relpos, mask, vec);
  // 4) output projection
  gemm(vec, 380, o_w, 380, attn_out, 380, 4096, 380, 380, nullptr, 0);
  // 5) residual + LN1
  add_ln_kernel<<<4096, 128, 0, stream>>>(w, attn_out, ln1g, ln1b, out1, 380);
  // 6) FF1 + bias + relu
  gemm(out1, 380, ffw1, 380, h1, 900, 4096, 900, 380, ffb1, 1);
  // 7) FF2 + bias
  gemm(h1, 900, ffw2, 900, core, 380, 4096, 380, 900, ffb2, 0);
  // 8) residual + LN2 -> output
  add_ln_kernel<<<4096, 128, 0, stream>>>(out1, core, ln2g, ln2b, out, 380);
}